// EfficientMoEMLPBlock_39127152067306
// MI455X (gfx1250) — compile-verified
//
#include <hip/hip_runtime.h>
#include <cstdint>
#include <cstddef>

// ---------------------------------------------------------------------------
// EfficientMoEMLPBlock for MI455X (gfx1250, wave32, WMMA + TDM)
//   x  [4,4096,1024] f32 ; Wr [48,1024] ; W1 [48,4,1024,1024] ; b1 [48,4,1024]
//   W2 [48,4,1024,256]   ; b2 [48,4,256]
//   cap = 512 tokens/expert (expert-choice routing)
// ---------------------------------------------------------------------------

typedef __attribute__((ext_vector_type(2))) float v2f;
typedef __attribute__((ext_vector_type(8))) float v8f;
typedef __attribute__((ext_vector_type(4))) unsigned int u32x4;
typedef __attribute__((ext_vector_type(8))) int i32x8;
typedef __attribute__((ext_vector_type(4))) int i32x4;

#define T_TOK 16384
#define DDIM  1024
#define NE    48
#define NS    4
#define MDIM  1024
#define KDIM  256
#define CAP   512
#define ROWS  32             // tokens per workgroup tile
#define NTILE (CAP / ROWS)   // 16
#define XPAD  1028           // +4 floats -> 4-bank shift per row (conflict-free cols)
#define MC    128            // m-chunk for fused GEMM1->GELU->GEMM2
#define HPAD  132            // MC + 4
#define W2PAD 260            // KDIM + 4 (TDM pad feature writes this stride)

// ---------------------------------------------------------------------------
// Kernel 1: router logits, stored transposed  logitsT[e][t]
// ---------------------------------------------------------------------------
__global__ __launch_bounds__(192) void router_logits(
    const float* __restrict__ x, const float* __restrict__ Wr,
    float* __restrict__ logitsT) {
  __shared__ float xs[4][DDIM];
  const int t0 = blockIdx.x * 4;
  for (int i = threadIdx.x; i < 4 * DDIM; i += 192)
    xs[i >> 10][i & 1023] = x[(size_t)(t0 + (i >> 10)) * DDIM + (i & 1023)];
  __syncthreads();
  const int tl = threadIdx.x / NE;   // 0..3
  const int e  = threadIdx.x % NE;   // 0..47
  const float* wr = Wr + (size_t)e * DDIM;
  const float* xr = xs[tl];
  float acc = 0.f;
#pragma unroll 8
  for (int d = 0; d < DDIM; ++d) acc += xr[d] * wr[d];
  logitsT[(size_t)e * T_TOK + t0 + tl] = acc;
}

// ---------------------------------------------------------------------------
// Kernel 2: per-expert exact top-512 (bitonic sort of 16384 64-bit keys in
// LDS — 128 KB, exploiting CDNA5's 320 KB/WGP) + softmax over selections.
// Key = sortable(float) << 32 | ~index  -> descending sort == jax.lax.top_k
// (ties resolved to lower index).
// ---------------------------------------------------------------------------
__global__ __launch_bounds__(1024) void topk_softmax(
    const float* __restrict__ logitsT, int* __restrict__ idx,
    float* __restrict__ wgt) {
  __shared__ unsigned long long keys[T_TOK];   // 128 KB
  __shared__ float red[CAP];
  const int e = blockIdx.x;
  const int tid = threadIdx.x;
  const float* lg = logitsT + (size_t)e * T_TOK;

  for (int i = tid; i < T_TOK; i += 1024) {
    unsigned b = __float_as_uint(lg[i]);
    unsigned k = (b & 0x80000000u) ? ~b : (b | 0x80000000u);
    keys[i] = ((unsigned long long)k << 32) | (unsigned)(~i);
  }
  __syncthreads();

  for (int k = 2; k <= T_TOK; k <<= 1) {
    for (int j = k >> 1; j > 0; j >>= 1) {
      for (int i = tid; i < T_TOK; i += 1024) {
        const int ixj = i ^ j;
        if (ixj > i) {
          unsigned long long a = keys[i], c = keys[ixj];
          const bool descBlock = ((i & k) == 0);
          if (descBlock ? (a < c) : (a > c)) { keys[i] = c; keys[ixj] = a; }
        }
      }
      __syncthreads();
    }
  }

  const int tok0 = (int)(~(unsigned)keys[0]);
  const float vmax = lg[tok0];
  int tok = 0; float ex = 0.f;
  if (tid < CAP) {
    tok = (int)(~(unsigned)keys[tid]);
    ex = __expf(lg[tok] - vmax);
    red[tid] = ex;
  }
  __syncthreads();
  for (int s = CAP / 2; s > 0; s >>= 1) {
    if (tid < s) red[tid] += red[tid + s];
    __syncthreads();
  }
  if (tid < CAP) {
    idx[e * CAP + tid] = tok;
    wgt[e * CAP + tid] = ex / red[0];
  }
}

// ---------------------------------------------------------------------------
// Kernel 3: zero output (scatter-add target)
// ---------------------------------------------------------------------------
__global__ void zero_out(float4* __restrict__ p, int n4) {
  int i = blockIdx.x * 256 + threadIdx.x;
  float4 z; z.x = 0.f; z.y = 0.f; z.z = 0.f; z.w = 0.f;
  if (i < n4) p[i] = z;
}

// ---------------------------------------------------------------------------
// TDM: async DMA of one W2 m-chunk [MC=128 rows x KDIM=256 cols, f32] into
// LDS with hardware row padding (+4 DWORDs every 256 DWORDs -> stride 260).
// Descriptor per CDNA5 ISA (2D tensor, groups 2/3 = 0).
// This toolchain exposes the 6-arg builtin:
//   (u32x4 g0, i32x8 g1, i32x4 g2, i32x4 g3, i32x8 g4, i32 cpol)
// ---------------------------------------------------------------------------
__device__ __forceinline__ void tdm_load_w2_chunk(const float* gptr,
                                                  unsigned lds_off) {
  const unsigned long long ga = (unsigned long long)(uintptr_t)gptr;
  u32x4 g0;
  g0[0] = 1u;                                           // count=1, no gather
  g0[1] = lds_off;                                      // lds_addr (bytes)
  g0[2] = (unsigned)(ga & 0xFFFFFFFFu);                 // global_addr[31:0]
  g0[3] = (unsigned)((ga >> 32) & 0x01FFFFFFu)          // global_addr[56:32]
          | (2u << 30);                                 // type=2 (image)
  i32x8 g1;
  g1[0] = (int)((2u << 16)        // data_size = 4 bytes
                | (1u << 20)      // pad_enable
                | (7u << 22)      // pad_interval = 256 DWORDs
                | (3u << 25));    // pad_amount  = 4 DWORDs
  g1[1] = (int)(256u << 16);      // tensor_dim0[15:0] = 256
  g1[2] = (int)(1024u << 16);     // tensor_dim0 hi=0 | tensor_dim1[15:0]=1024
  g1[3] = (int)(256u << 16);      // tensor_dim1 hi=0 | tile_dim0 = 256
  g1[4] = MC;                     // tile_dim1 = 128 (tile_dim2 = 0)
  g1[5] = 256;                    // tensor_dim0_stride = 256 elements
  g1[6] = 0;
  g1[7] = 0;
  i32x4 z4; z4[0] = 0; z4[1] = 0; z4[2] = 0; z4[3] = 0;
  i32x8 z8;
#pragma unroll
  for (int i = 0; i < 8; ++i) z8[i] = 0;
  __builtin_amdgcn_tensor_load_to_lds(g0, g1, z4, z4, z8, 0);
}

// ---------------------------------------------------------------------------
// Kernel 4: fused gather -> GEMM1(fp32 WMMA) -> +b1 -> GELU(erf)
//           -> GEMM2(fp32 WMMA, B from TDM-loaded LDS) -> +b2 -> *route_w
//           -> atomic scatter-add.
// One workgroup per (expert, shard, 32-token tile); 8 waves.
//   GEMM1: wave w owns h cols [w*16, w*16+16) of the 128-wide chunk and both
//          16-row groups -> B fragment shared by 2 WMMAs.
//   GEMM2: wave w owns y col-groups {2w, 2w+1} x both row groups (4 accums).
// ---------------------------------------------------------------------------
__global__ __launch_bounds__(256) void moe_mlp(
    const float* __restrict__ x,  const float* __restrict__ W1,
    const float* __restrict__ b1, const float* __restrict__ W2,
    const float* __restrict__ b2, const int* __restrict__ idx,
    const float* __restrict__ wgt, float* __restrict__ out) {
  __shared__ float Xs[ROWS * XPAD];    // 131,584 B  gathered activations
  __shared__ float Hs[ROWS * HPAD];    //  16,896 B  GELU'd h chunk
  __shared__ float W2s[MC * W2PAD];    // 133,120 B  TDM-loaded W2 chunk
  __shared__ int   tokS[ROWS];
  __shared__ float wS[ROWS];

  const int b    = blockIdx.x;
  const int tile = b & (NTILE - 1);
  const int s    = (b >> 4) & (NS - 1);
  const int e    = b >> 6;
  const int tid  = threadIdx.x;
  const int wave = tid >> 5;
  const int lane = tid & 31;
  const int lmod = lane & 15;     // column within 16x16 tile
  const int lhalf = lane >> 4;    // half-wave: K pairs {0,1} vs {2,3}

  if (tid < ROWS) {
    tokS[tid] = idx[e * CAP + tile * ROWS + tid];
    wS[tid]   = wgt[e * CAP + tile * ROWS + tid];
  }
  __syncthreads();
  // gather 32 token rows into LDS (coalesced)
  for (int i = tid; i < ROWS * DDIM; i += 256) {
    const int r = i >> 10, d = i & 1023;
    Xs[r * XPAD + d] = x[(size_t)tokS[r] * DDIM + d];
  }

  const size_t w1base = ((size_t)e * NS + s) * (size_t)DDIM * MDIM;
  const float* w2chunk0 = W2 + ((size_t)e * NS + s) * (size_t)MDIM * KDIM;
  const unsigned w2s_lds = (unsigned)(uintptr_t)&W2s[0];

  v8f yacc[2][2];   // [row group][col pair]
#pragma unroll
  for (int rg = 0; rg < 2; ++rg)
#pragma unroll
    for (int t = 0; t < 2; ++t)
#pragma unroll
      for (int v = 0; v < 8; ++v) yacc[rg][t][v] = 0.f;

  __syncthreads();   // Xs ready

  for (int mbase = 0; mbase < MDIM; mbase += MC) {
    // kick off async DMA of this chunk's W2 block; overlaps all of GEMM1
    if (tid == 0) tdm_load_w2_chunk(w2chunk0 + (size_t)mbase * KDIM, w2s_lds);

    // prefetch a slice of the next W1 m-chunk into L2
    if (mbase + MC < MDIM) {
      const float* pf = W1 + w1base + (size_t)tid * MDIM + (mbase + MC);
      __builtin_prefetch(pf, 0, 1);
      __builtin_prefetch(pf + 256 * MDIM, 0, 1);
      __builtin_prefetch(pf + 512 * MDIM, 0, 1);
      __builtin_prefetch(pf + 768 * MDIM, 0, 1);
    }

    // ---- GEMM1: h[32 x 16] = X[32 x 1024] * W1[1024 x 16] (per wave) ------
    v8f hacc0, hacc1;
#pragma unroll
    for (int v = 0; v < 8; ++v) { hacc0[v] = 0.f; hacc1[v] = 0.f; }

    const float* w1p  = W1 + w1base + (size_t)(lhalf * 2) * MDIM
                        + (mbase + wave * 16 + lmod);
    const float* xsA0 = Xs + lmod * XPAD + lhalf * 2;          // rows 0..15
    const float* xsA1 = Xs + (16 + lmod) * XPAD + lhalf * 2;   // rows 16..31
#pragma unroll 4
    for (int d = 0; d < DDIM; d += 4) {
      v2f a0 = *(const v2f*)xsA0;            // ds_load_b64, conflict-free
      v2f a1 = *(const v2f*)xsA1;
      v2f bb; bb.x = w1p[0]; bb.y = w1p[MDIM];
      hacc0 = __builtin_amdgcn_wmma_f32_16x16x4_f32(
          false, a0, false, bb, (short)0, hacc0, false, false);
      hacc1 = __builtin_amdgcn_wmma_f32_16x16x4_f32(
          false, a1, false, bb, (short)0, hacc1, false, false);
      xsA0 += 4; xsA1 += 4;
      w1p += 4 * MDIM;
    }

    // bias + exact GELU, stash chunk in LDS as GEMM2 A-operand
    const float bias = b1[(e * NS + s) * MDIM + mbase + wave * 16 + lmod];
#pragma unroll
    for (int v = 0; v < 8; ++v) {
      float h0 = hacc0[v] + bias;
      float h1 = hacc1[v] + bias;
      h0 = 0.5f * h0 * (1.f + erff(h0 * 0.70710678118654752f));
      h1 = 0.5f * h1 * (1.f + erff(h1 * 0.70710678118654752f));
      Hs[(lhalf * 8 + v) * HPAD + wave * 16 + lmod] = h0;
      Hs[(16 + lhalf * 8 + v) * HPAD + wave * 16 + lmod] = h1;
    }
    if (tid == 0) __builtin_amdgcn_s_wait_tensorcnt(0);  // W2 chunk landed
    __syncthreads();                                      // Hs + W2s visible

    // ---- GEMM2: y[32 x 32] += h[32 x 128] * W2s[128 x 32] (per wave) ------
    const float* hsA0 = Hs + lmod * HPAD + lhalf * 2;
    const float* hsA1 = Hs + (16 + lmod) * HPAD + lhalf * 2;
    const int col0 = (2 * wave) * 16 + lmod;
    const int col1 = col0 + 16;
#pragma unroll 2
    for (int mk = 0; mk < MC; mk += 4) {
      v2f a0 = *(const v2f*)(hsA0 + mk);
      v2f a1 = *(const v2f*)(hsA1 + mk);
      const float* wrow = W2s + (mk + lhalf * 2) * W2PAD;
      v2f b0; b0.x = wrow[col0]; b0.y = wrow[W2PAD + col0];
      v2f b1v; b1v.x = wrow[col1]; b1v.y = wrow[W2PAD + col1];
      yacc[0][0] = __builtin_amdgcn_wmma_f32_16x16x4_f32(
          false, a0, false, b0, (short)0, yacc[0][0], false, false);
      yacc[1][0] = __builtin_amdgcn_wmma_f32_16x16x4_f32(
          false, a1, false, b0, (short)0, yacc[1][0], false, false);
      yacc[0][1] = __builtin_amdgcn_wmma_f32_16x16x4_f32(
          false, a0, false, b1v, (short)0, yacc[0][1], false, false);
      yacc[1][1] = __builtin_amdgcn_wmma_f32_16x16x4_f32(
          false, a1, false, b1v, (short)0, yacc[1][1], false, false);
    }
    __syncthreads();   // Hs/W2s reused next chunk
  }

  // ---- epilogue: +b2, * routing weight, scatter-add ------------------------
#pragma unroll
  for (int t = 0; t < 2; ++t) {
    const int col = (2 * wave + t) * 16 + lmod;
    const float b2v = b2[(e * NS + s) * KDIM + col];
#pragma unroll
    for (int rg = 0; rg < 2; ++rg)
#pragma unroll
      for (int v = 0; v < 8; ++v) {
        const int row = rg * 16 + lhalf * 8 + v;
        const float val = (yacc[rg][t][v] + b2v) * wS[row];
        atomicAdd(&out[(size_t)tokS[row] * DDIM + s * KDIM + col], val);
      }
  }
}

// ---------------------------------------------------------------------------
extern "C" void kernel_launch(void* const* d_in, const int* in_sizes, int n_in,
                              void* d_out, int out_size, void* d_ws,
                              size_t ws_size, hipStream_t stream) {
  const float* x  = (const float*)d_in[0];
  const float* Wr = (const float*)d_in[1];
  const float* W1 = (const float*)d_in[2];
  const float* b1 = (const float*)d_in[3];
  const float* W2 = (const float*)d_in[4];
  const float* b2 = (const float*)d_in[5];
  float* out = (float*)d_out;

  char* ws = (char*)d_ws;
  float* logitsT = (float*)ws;                                   // 3 MB
  int*   idx = (int*)(ws + sizeof(float) * NE * T_TOK);          // 96 KB
  float* wgt = (float*)(ws + sizeof(float) * NE * T_TOK
                           + sizeof(int) * NE * CAP);            // 96 KB

  const int n4 = out_size / 4;
  zero_out<<<(n4 + 255) / 256, 256, 0, stream>>>((float4*)out, n4);
  router_logits<<<T_TOK / 4, 192, 0, stream>>>(x, Wr, logitsT);
  topk_softmax<<<NE, 1024, 0, stream>>>(logitsT, idx, wgt);
  moe_mlp<<<NE * NS * NTILE, 256, 0, stream>>>(x, W1, b1, W2, b2, idx, wgt,
                                               out);
}